// BatchedExpertDispatch_63668595196397
// MI455X (gfx1250) — compile-verified
//
#include <hip/hip_runtime.h>
#include <math.h>

typedef __attribute__((ext_vector_type(2))) float v2f;
typedef __attribute__((ext_vector_type(4))) float v4f;
typedef __attribute__((ext_vector_type(8))) float v8f;

#define NUM_EXPERTS 64
#define KCHUNKS 16  // 64 experts / K=4 per WMMA step

// candidate comparison matching jax.lax.top_k tie-breaking (lower index wins ties)
__device__ __forceinline__ bool better(float av, int ai, float bv, int bi) {
    return (av > bv) || (av == bv && ai < bi);
}

// One wave (32 lanes) handles 16 tokens x 64 experts.
// Lane L (<16) holds token L, experts congruent {0,1} mod 4.
// Lane L (>=16) holds token L-16, experts congruent {2,3} mod 4.
// This is exactly the V_WMMA_F32_16X16X4_F32 A-operand layout per K-chunk.
__global__ __launch_bounds__(256) void moe_router_kernel(
    const float* __restrict__ logits,  // [N, 64]
    int*   __restrict__ idx_out,       // [N, 2]
    float* __restrict__ w_out,         // [N, 2]
    int N)
{
    const int lane   = threadIdx.x & 31;
    const int wave   = threadIdx.x >> 5;
    const int tokBase = blockIdx.x * 128 + wave * 16;
    const int m      = lane & 15;                 // token-in-tile (matrix row M)
    const int token  = tokBase + m;
    const int eoff   = (lane & 16) ? 2 : 0;       // expert sub-offset inside K-chunk

    // ---- load this lane's 32 logits as 16 aligned float2 (adjacent expert pairs)
    v2f a[KCHUNKS];
    const float* row = logits + (size_t)token * NUM_EXPERTS + eoff;
    #pragma unroll
    for (int kc = 0; kc < KCHUNKS; ++kc)
        a[kc] = *reinterpret_cast<const v2f*>(row + kc * 4);

    // ---- row max (softmax stability): local max, then merge lane pair L <-> L^16
    float mx = -__builtin_huge_valf();
    #pragma unroll
    for (int kc = 0; kc < KCHUNKS; ++kc)
        mx = fmaxf(mx, fmaxf(a[kc].x, a[kc].y));
    mx = fmaxf(mx, __shfl_xor(mx, 16, 32));

    // ---- exponentials (kept in the WMMA A layout)
    #pragma unroll
    for (int kc = 0; kc < KCHUNKS; ++kc) {
        a[kc].x = __expf(a[kc].x - mx);
        a[kc].y = __expf(a[kc].y - mx);
    }

    // ---- softmax denominator Z via matrix unit: D = sum_k A(16x4) * ones(4x16), acc over 16 chunks
    v8f acc = {0.f, 0.f, 0.f, 0.f, 0.f, 0.f, 0.f, 0.f};
    v2f ones; ones.x = 1.0f; ones.y = 1.0f;
    #pragma unroll
    for (int kc = 0; kc < KCHUNKS; ++kc) {
        acc = __builtin_amdgcn_wmma_f32_16x16x4_f32(
            /*neg_a=*/false, a[kc], /*neg_b=*/false, ones,
            /*c_mod=*/(short)0, acc, /*reuse_a=*/false, /*reuse_b=*/false);
    }

    // ---- extract Z_m for this lane's token from the C/D layout:
    // VGPR r, lanes 0-15 hold M=r, lanes 16-31 hold M=r+8, replicated across N.
    {
        // nothing
    }
    const int r = lane & 7;
    float local =
        (r == 0) ? acc[0] : (r == 1) ? acc[1] : (r == 2) ? acc[2] : (r == 3) ? acc[3] :
        (r == 4) ? acc[4] : (r == 5) ? acc[5] : (r == 6) ? acc[6] : acc[7];
    float partner = __shfl_xor(local, 16, 32);
    // token's Z is in this half's registers iff (m<8) == (lane<16)
    bool same_half = (((lane & 8) == 0) == ((lane & 16) == 0));
    float Z = same_half ? local : partner;

    // ---- per-lane top-2 over this lane's 32 exp values (indices ascend -> stable ties)
    float bv0 = -__builtin_huge_valf(), bv1 = -__builtin_huge_valf();
    int   bi0 = 0x7fffffff,             bi1 = 0x7fffffff;
    #pragma unroll
    for (int kc = 0; kc < KCHUNKS; ++kc) {
        #pragma unroll
        for (int j = 0; j < 2; ++j) {
            float v = (j == 0) ? a[kc].x : a[kc].y;
            int   e = 4 * kc + eoff + j;
            if (better(v, e, bv0, bi0)) { bv1 = bv0; bi1 = bi0; bv0 = v; bi0 = e; }
            else if (better(v, e, bv1, bi1)) { bv1 = v; bi1 = e; }
        }
    }

    // ---- merge with partner lane's top-2 (lane pair covers the full 64-expert row)
    float pv0 = __shfl_xor(bv0, 16, 32);
    float pv1 = __shfl_xor(bv1, 16, 32);
    int   pi0 = __shfl_xor(bi0, 16, 32);
    int   pi1 = __shfl_xor(bi1, 16, 32);

    float tv0, tv1; int ti0, ti1;
    if (better(bv0, bi0, pv0, pi0)) {
        tv0 = bv0; ti0 = bi0;
        if (better(pv0, pi0, bv1, bi1)) { tv1 = pv0; ti1 = pi0; }
        else                            { tv1 = bv1; ti1 = bi1; }
    } else {
        tv0 = pv0; ti0 = pi0;
        if (better(bv0, bi0, pv1, pi1)) { tv1 = bv0; ti1 = bi0; }
        else                            { tv1 = pv1; ti1 = pi1; }
    }

    // ---- follow reference numerics: probs = exp/Z, then normalize the top-2
    float p0 = tv0 / Z;
    float p1 = tv1 / Z;
    float s  = p0 + p1;
    float w0 = p0 / s;
    float w1 = p1 / s;

    if (lane < 16) {
        idx_out[2 * token + 0] = ti0;
        idx_out[2 * token + 1] = ti1;
        w_out[2 * token + 0] = w0;
        w_out[2 * token + 1] = w1;
    }
}

// combined[n,:] = x[n,:]*w0[n] + x[n,:]*w1[n]  (identity experts collapse the
// permutation round trip).  Pure 256 MB stream -> 128-bit non-temporal access.
__global__ __launch_bounds__(256) void moe_combine_kernel(
    const float* __restrict__ x,
    const float* __restrict__ w,   // [N,2] normalized routing weights
    float* __restrict__ out,
    int H4, long long total4)
{
    long long i = (long long)blockIdx.x * blockDim.x + threadIdx.x;
    if (i >= total4) return;
    int n = (int)(i / H4);
    const v4f* xp = reinterpret_cast<const v4f*>(x) + i;
    v4f v = __builtin_nontemporal_load(xp);
    float w0 = w[2 * n + 0];
    float w1 = w[2 * n + 1];
    v4f res = v * w0 + v * w1;   // elementwise, same op order as reference sum over k
    __builtin_nontemporal_store(res, reinterpret_cast<v4f*>(out) + i);
}

extern "C" void kernel_launch(void* const* d_in, const int* in_sizes, int n_in,
                              void* d_out, int out_size, void* d_ws, size_t ws_size,
                              hipStream_t stream) {
    (void)n_in; (void)d_ws; (void)ws_size; (void)out_size;
    const float* hidden = (const float*)d_in[0];   // [N, H] f32
    const float* logits = (const float*)d_in[1];   // [N, 64] f32
    // d_in[2] is top_k == 2 (layout below assumes k=2)

    const int N = in_sizes[1] / NUM_EXPERTS;       // 8192
    const int H = in_sizes[0] / N;                 // 4096

    // output layout: combined [N*H] f32 | expert_indices [N*2] i32 | routing_weights [N*2] f32
    float* out_combined = (float*)d_out;
    int*   out_idx      = (int*)(out_combined + (size_t)N * H);
    float* out_w        = out_combined + (size_t)N * H + 2 * (size_t)N;

    // router: 1 wave per 16 tokens, 8 waves per block -> 128 tokens/block
    moe_router_kernel<<<N / 128, 256, 0, stream>>>(logits, out_idx, out_w, N);

    // combine: one float4 per thread
    long long total4 = (long long)N * H / 4;
    int H4 = H / 4;
    int blocks = (int)((total4 + 255) / 256);
    moe_combine_kernel<<<blocks, 256, 0, stream>>>(hidden, out_w, out_combined, H4, total4);
}